// GraphTransformer_82248623719126
// MI455X (gfx1250) — compile-verified
//
#include <hip/hip_runtime.h>
#include <hip/hip_bf16.h>
#include <cstdint>

typedef __attribute__((ext_vector_type(2))) float    v2f;
typedef __attribute__((ext_vector_type(8))) float    v8f;
typedef __attribute__((ext_vector_type(4))) unsigned v4u;
typedef __attribute__((ext_vector_type(8))) int      v8i;
typedef __attribute__((ext_vector_type(4))) int      v4i;

#define NEG_SLOPE 0.2f
#define LDSK 132   // 128 + 4 DWORD pad per row (TDM pad_interval=128dw, pad_amount=4dw)

// ---------------------------------------------------------------------------
// Monotonic float <-> uint key for atomic float max via atomicMax(uint).
// key 0u decodes below any real float key (acts as -inf init).
// ---------------------------------------------------------------------------
__device__ __forceinline__ unsigned fkey(float x) {
  unsigned b = __float_as_uint(x);
  return (b & 0x80000000u) ? ~b : (b | 0x80000000u);
}
__device__ __forceinline__ float fkey_inv(unsigned k) {
  unsigned b = (k & 0x80000000u) ? (k & 0x7FFFFFFFu) : ~k;
  return __uint_as_float(b);
}

// ---------------------------------------------------------------------------
// fp32 GEMM, Y = X @ W.  X:[M,K], W:[K,Nw], Y:[M,Nw]; M,Nw multiples of 16,
// K == 128 here.  One 256-thread block (8 waves) per 16-row M tile:
//   - TDM (tensor_load_to_lds) stages the 16xK A tile into LDS once, with
//     4-DWORD padding every 128 DWORDs -> LDS row stride 132 floats so the
//     16 A-fragment lanes hit distinct banks (ds_load_b64, 4*m bank spacing).
//   - Each wave computes 16x16 output tiles via V_WMMA_F32_16X16X4_F32,
//     striding over N tiles (A tile reused 8-10x from LDS).
// WMMA f32 layouts (ISA 7.12.2):
//   A 16x4 : lanes 0-15 -> K=0,1 ; lanes 16-31 -> K=2,3 (per 4-step)
//   B 4x16 : VGPR0 rows {0,2}, VGPR1 rows {1,3}, N striped over lane%16
//   C/D    : VGPR v, lanes 0-15 -> M=v, lanes 16-31 -> M=v+8
// ---------------------------------------------------------------------------
__global__ void __launch_bounds__(256)
wmma_gemm_f32_lds(const float* __restrict__ X, const float* __restrict__ W,
                  float* __restrict__ Y, int M, int K, int Nw) {
  __shared__ float As[16 * LDSK];

  const int tm   = blockIdx.x;
  const int wave = threadIdx.x >> 5;
  const int lane = threadIdx.x & 31;
  const int l16  = lane & 15;
  const int half = lane >> 4;
  const int kk   = half * 2;

  // ---- TDM: DMA X[tm*16 .. tm*16+16) x [0,K) into LDS (wave 0 issues) ----
  if (threadIdx.x < 32) {
    const unsigned lds0 = (unsigned)(size_t)(const void*)As;  // LDS byte offset
    const unsigned long long ga =
        (unsigned long long)(size_t)(const void*)(X + (size_t)tm * 16 * K);
    v4u g0;
    g0[0] = 1u;                                   // count=1, user mode, no gather
    g0[1] = lds0;                                 // D#.lds_addr
    g0[2] = (unsigned)(ga & 0xFFFFFFFFull);       // global_addr[31:0]
    g0[3] = (unsigned)((ga >> 32) & 0x01FFFFFFull) | 0x80000000u; // [56:32]|type=2
    v8i g1;
    g1[0] = (int)0x07920000u;  // data_size=4B, pad_enable, interval=128dw, amount=4dw
    g1[1] = (int)((unsigned)(K & 0xFFFF) << 16);          // tensor_dim0[15:0]
    g1[2] = (int)(((unsigned)K >> 16) | (16u << 16));     // dim0[31:16] | dim1[15:0]
    g1[3] = (int)((unsigned)(K & 0xFFFF) << 16);          // dim1[31:16]=0 | tile_dim0
    g1[4] = 16;                                           // tile_dim1=16, tile_dim2=0
    g1[5] = K;                                            // tensor_dim0_stride[31:0]
    g1[6] = 0;                                            // stride0 hi | stride1 lo
    g1[7] = 0;                                            // stride1 hi
    v4i g2 = {0, 0, 0, 0};                                // 2D tensor: groups 2/3 unused
    v4i g3 = {0, 0, 0, 0};
    v8i g4 = {0, 0, 0, 0, 0, 0, 0, 0};                    // extra group (clang-23 form)
    __builtin_amdgcn_tensor_load_to_lds(g0, g1, g2, g3, g4, 0);
    __builtin_amdgcn_s_wait_tensorcnt(0);
  }
  __syncthreads();

  const int Ntiles = Nw >> 4;
  for (int tn = wave; tn < Ntiles; tn += 8) {
    const int ncol = tn * 16 + l16;
    v8f c = {};
#pragma unroll 4
    for (int k = 0; k < K; k += 4) {
      v2f a, b;
      a.x = As[l16 * LDSK + k + kk];                 // ds_load, bank-conflict free
      a.y = As[l16 * LDSK + k + kk + 1];
      b.x = W[(size_t)(k + kk) * Nw + ncol];
      b.y = W[(size_t)(k + kk + 1) * Nw + ncol];
      c = __builtin_amdgcn_wmma_f32_16x16x4_f32(
          /*neg_a=*/false, a, /*neg_b=*/false, b,
          /*c_mod=*/(short)0, c, /*reuse_a=*/false, /*reuse_b=*/false);
    }
#pragma unroll
    for (int v = 0; v < 8; ++v) {
      const int row = tm * 16 + v + half * 8;
      if (row < M) Y[(size_t)row * Nw + ncol] = c[v];
    }
  }
}

// ---------------------------------------------------------------------------
// Per-(node, head) attention scores: s_src[n,h] = <h[n,h,:], a_src[h,:]>
// ---------------------------------------------------------------------------
__global__ void node_scores(const float* __restrict__ h,
                            const float* __restrict__ a_src,
                            const float* __restrict__ a_dst,
                            float* __restrict__ s_src,
                            float* __restrict__ s_dst,
                            int N, int Hh, int C) {
  const int idx = blockIdx.x * blockDim.x + threadIdx.x;
  if (idx >= N * Hh) return;
  const int n = idx / Hh, hd = idx - n * Hh;
  const float* __restrict__ hp = h + (size_t)n * Hh * C + (size_t)hd * C;
  const float* __restrict__ as = a_src + (size_t)hd * C;
  const float* __restrict__ ad = a_dst + (size_t)hd * C;
  float ss = 0.f, sd = 0.f;
  for (int c = 0; c < C; ++c) {
    const float v = hp[c];
    ss += v * as[c];
    sd += v * ad[c];
  }
  s_src[idx] = ss;
  s_dst[idx] = sd;
}

__device__ __forceinline__ void edge_pair(const int* __restrict__ ei, int e,
                                          int E, int& src, int& dst) {
  if (e < E) { src = ei[e]; dst = ei[E + e]; }
  else       { src = dst = e - E; }      // implicit self-loop
}

// Pass A: per-dst, per-head running max of leaky_relu(s_src[src]+s_dst[dst])
__global__ void edge_max(const int* __restrict__ ei,
                         const float* __restrict__ s_src,
                         const float* __restrict__ s_dst,
                         unsigned* __restrict__ mkey,
                         int E, int Etot, int Hh) {
  const int id = blockIdx.x * blockDim.x + threadIdx.x;
  if (id >= Etot * Hh) return;
  const int e = id / Hh, hd = id - e * Hh;
  int src, dst; edge_pair(ei, e, E, src, dst);
  float x = s_src[src * Hh + hd] + s_dst[dst * Hh + hd];
  x = (x > 0.f) ? x : NEG_SLOPE * x;
  atomicMax(&mkey[dst * Hh + hd], fkey(x));
}

// Pass B: p = exp(e - m[dst]); store p per edge-head, atomic-sum into z[dst]
__global__ void edge_expsum(const int* __restrict__ ei,
                            const float* __restrict__ s_src,
                            const float* __restrict__ s_dst,
                            const unsigned* __restrict__ mkey,
                            float* __restrict__ z,
                            float* __restrict__ pbuf,
                            int E, int Etot, int Hh) {
  const int id = blockIdx.x * blockDim.x + threadIdx.x;
  if (id >= Etot * Hh) return;
  const int e = id / Hh, hd = id - e * Hh;
  int src, dst; edge_pair(ei, e, E, src, dst);
  float x = s_src[src * Hh + hd] + s_dst[dst * Hh + hd];
  x = (x > 0.f) ? x : NEG_SLOPE * x;
  const float m = fkey_inv(mkey[dst * Hh + hd]);
  const float p = expf(x - m);
  pbuf[id] = p;
  atomicAdd(&z[dst * Hh + hd], p);
}

// Pass C: out[dst] += (p/z[dst]) * h[src]. One block of F=Hh*C threads / edge.
__global__ void edge_aggregate(const int* __restrict__ ei,
                               const float* __restrict__ h,
                               const float* __restrict__ pbuf,
                               const float* __restrict__ z,
                               float* __restrict__ out,
                               int E, int Hh, int C) {
  const int e = blockIdx.x;
  const int t = threadIdx.x;             // 0..F-1
  int src, dst; edge_pair(ei, e, E, src, dst);
  const int hd = t / C;
  const float alpha = pbuf[(size_t)e * Hh + hd] / z[dst * Hh + hd];
  const int F = Hh * C;
  const float v = h[(size_t)src * F + t] * alpha;
  atomicAdd(&out[(size_t)dst * F + t], v);
}

// Layer-1 epilogue: x2 = elu(acc + b1)  (elu alpha = 1)
__global__ void post_layer1(const float* __restrict__ acc,
                            const float* __restrict__ b1,
                            float* __restrict__ x2, int N, int F) {
  const int idx = blockIdx.x * blockDim.x + threadIdx.x;
  if (idx >= N * F) return;
  const int f = idx % F;
  const float v = acc[idx] + b1[f];
  x2[idx] = (v > 0.f) ? v : (expf(v) - 1.f);
}

// Layer-2 epilogue: mean over heads + b2, then log_softmax over 40 classes
__global__ void post_layer2(const float* __restrict__ acc,
                            const float* __restrict__ b2,
                            float* __restrict__ out, int N, int Hh) {
  const int n = blockIdx.x * blockDim.x + threadIdx.x;
  if (n >= N) return;
  const int OUTC = 40;
  float vals[40];
  float mx = -3.402823466e38f;
  const float invH = 1.f / (float)Hh;
  for (int o = 0; o < OUTC; ++o) {
    float s = 0.f;
    for (int hd = 0; hd < Hh; ++hd)
      s += acc[(size_t)n * Hh * OUTC + (size_t)hd * OUTC + o];
    const float v = s * invH + b2[o];
    vals[o] = v;
    mx = fmaxf(mx, v);
  }
  float zs = 0.f;
  for (int o = 0; o < OUTC; ++o) zs += expf(vals[o] - mx);
  const float lz = mx + logf(zs);
  for (int o = 0; o < OUTC; ++o) out[(size_t)n * OUTC + o] = vals[o] - lz;
}

// ---------------------------------------------------------------------------
extern "C" void kernel_launch(void* const* d_in, const int* in_sizes, int n_in,
                              void* d_out, int out_size, void* d_ws, size_t ws_size,
                              hipStream_t stream) {
  const float* x   = (const float*)d_in[0];
  const int*   ei  = (const int*)  d_in[1];
  const float* W1  = (const float*)d_in[2];
  const float* as1 = (const float*)d_in[3];
  const float* ad1 = (const float*)d_in[4];
  const float* b1  = (const float*)d_in[5];
  const float* W2  = (const float*)d_in[6];
  const float* as2 = (const float*)d_in[7];
  const float* ad2 = (const float*)d_in[8];
  const float* b2  = (const float*)d_in[9];
  float* out = (float*)d_out;

  const int IN = 128, Hh = 4, HID = 32, OUTC = 40;
  const int F1 = Hh * HID;               // 128
  const int F2 = Hh * OUTC;              // 160
  const int N  = in_sizes[0] / IN;       // 100000
  const int E  = in_sizes[1] / 2;        // 1.6M
  const int Etot = E + N;                // + self loops

  // --- workspace carve (~213 MB) ---
  float*    A     = (float*)d_ws;                  // N*F1   : h1, later x2
  float*    B     = A + (size_t)N * F1;            // N*F2   : acc1 (F1 used), then h2
  float*    Cb    = B + (size_t)N * F2;            // N*F2   : acc2
  float*    ssrc  = Cb + (size_t)N * F2;           // N*Hh
  float*    sdst  = ssrc + (size_t)N * Hh;         // N*Hh
  unsigned* mkey  = (unsigned*)(sdst + (size_t)N * Hh);   // N*Hh
  float*    z     = (float*)(mkey + (size_t)N * Hh);      // N*Hh
  float*    pbuf  = z + (size_t)N * Hh;            // Etot*Hh

  const int THREADS = 256;
  const int nhBlocks = (N * Hh + THREADS - 1) / THREADS;
  const int ehBlocks = (Etot * Hh + THREADS - 1) / THREADS;
  const int Mtiles = (N + 15) / 16;

  // ======================= Layer 1 =======================
  wmma_gemm_f32_lds<<<Mtiles, 256, 0, stream>>>(x, W1, A, N, IN, F1);
  node_scores<<<nhBlocks, THREADS, 0, stream>>>(A, as1, ad1, ssrc, sdst, N, Hh, HID);

  (void)hipMemsetAsync(mkey, 0, (size_t)N * Hh * sizeof(unsigned), stream); // key(-inf)
  (void)hipMemsetAsync(z,    0, (size_t)N * Hh * sizeof(float),    stream);
  (void)hipMemsetAsync(B,    0, (size_t)N * F1 * sizeof(float),    stream);

  edge_max   <<<ehBlocks, THREADS, 0, stream>>>(ei, ssrc, sdst, mkey, E, Etot, Hh);
  edge_expsum<<<ehBlocks, THREADS, 0, stream>>>(ei, ssrc, sdst, mkey, z, pbuf, E, Etot, Hh);
  edge_aggregate<<<Etot, F1, 0, stream>>>(ei, A, pbuf, z, B, E, Hh, HID);
  post_layer1<<<(N * F1 + THREADS - 1) / THREADS, THREADS, 0, stream>>>(B, b1, A, N, F1);

  // ======================= Layer 2 =======================
  wmma_gemm_f32_lds<<<Mtiles, 256, 0, stream>>>(A, W2, B, N, F1, F2);
  node_scores<<<nhBlocks, THREADS, 0, stream>>>(B, as2, ad2, ssrc, sdst, N, Hh, OUTC);

  (void)hipMemsetAsync(mkey, 0, (size_t)N * Hh * sizeof(unsigned), stream);
  (void)hipMemsetAsync(z,    0, (size_t)N * Hh * sizeof(float),    stream);
  (void)hipMemsetAsync(Cb,   0, (size_t)N * F2 * sizeof(float),    stream);

  edge_max   <<<ehBlocks, THREADS, 0, stream>>>(ei, ssrc, sdst, mkey, E, Etot, Hh);
  edge_expsum<<<ehBlocks, THREADS, 0, stream>>>(ei, ssrc, sdst, mkey, z, pbuf, E, Etot, Hh);
  edge_aggregate<<<Etot, F2, 0, stream>>>(ei, B, pbuf, z, Cb, E, Hh, OUTC);
  post_layer2<<<(N + THREADS - 1) / THREADS, THREADS, 0, stream>>>(Cb, b2, out, N, Hh);
}